// GraphAttentionLayerEdge_45028437131376
// MI455X (gfx1250) — compile-verified
//
#include <hip/hip_runtime.h>
#include <hip/hip_bf16.h>

// ---- problem constants (from reference) ----
#define N_NODES 80000
#define N_EDGES 1280000
#define D_NODE  128
#define D_EDGE  32
#define D_HID   64
#define NEG_SLOPE 0.2f

typedef __attribute__((ext_vector_type(2))) float v2f;
typedef __attribute__((ext_vector_type(8))) float v8f;

// ---------------------------------------------------------------------------
// k_prep: fold the edge-feature path into a single 32-vector.
//   ve[k] = sum_h ew_w[k][h] * a_w[128+h]     (edge_h only feeds the score dot)
//   c     = sum_h ew_b[h]   * a_w[128+h] + a_b
// ---------------------------------------------------------------------------
__global__ void k_prep(const float* __restrict__ eww, const float* __restrict__ ewb,
                       const float* __restrict__ aw,  const float* __restrict__ ab,
                       float* __restrict__ ve, float* __restrict__ cptr) {
    int k = threadIdx.x;
    if (k < D_EDGE) {
        float s = 0.f;
        for (int h = 0; h < D_HID; ++h) s += eww[k * D_HID + h] * aw[2 * D_HID + h];
        ve[k] = s;
    }
    if (k == D_EDGE) {
        float s = ab[0];
        for (int h = 0; h < D_HID; ++h) s += ewb[h] * aw[2 * D_HID + h];
        cptr[0] = s;
    }
}

// ---------------------------------------------------------------------------
// k_init: zero output accumulator + segment sums; seed segment-max keys with
// the ordered-uint encoding of -inf (msb-flip trick: unsigned order == float
// order).
// ---------------------------------------------------------------------------
__global__ void k_init(float* __restrict__ out, float* __restrict__ segsum,
                       unsigned* __restrict__ keys) {
    int i = blockIdx.x * blockDim.x + threadIdx.x;
    if (i < N_NODES * D_HID) out[i] = 0.f;
    if (i < N_NODES) { segsum[i] = 0.f; keys[i] = 0x007FFFFFu; /* key(-inf) */ }
}

// ---------------------------------------------------------------------------
// k_node_gemm: h = node_features @ w_w + w_b  via V_WMMA_F32_16X16X4_F32.
// One wave per 16-row node tile, producing ALL 64 output columns (4 C frags):
// each A fragment load feeds 4 WMMAs, and node_features is read exactly once.
// w_w staged in LDS (32 KB).
// A frag (16x4): lanes 0-15 hold M=lane, VGPR0/1 = K {0,1}; lanes 16-31 = K {2,3}.
// B frag (4x16): VGPRv, lanes 0-15 -> K=v, N=lane; lanes 16-31 -> K=v+2.
// C (16x16):     VGPRv, lanes 0-15 -> M=v, N=lane; lanes 16-31 -> M=v+8.
// ---------------------------------------------------------------------------
__global__ __launch_bounds__(256) void k_node_gemm(const float* __restrict__ nf,
                                                   const float* __restrict__ ww,
                                                   const float* __restrict__ wb,
                                                   float* __restrict__ h) {
    __shared__ float sW[D_NODE * D_HID];                 // 128x64 f32 = 32 KB
    for (int i = threadIdx.x; i < D_NODE * D_HID; i += 256) sW[i] = ww[i];
    __syncthreads();

    const int wid  = threadIdx.x >> 5;
    const int lane = threadIdx.x & 31;
    const int tile = blockIdx.x * 8 + wid;               // 16-row node tile (5000)
    const int lo   = lane & 15;
    const int hi   = lane >> 4;                          // 0 or 1
    const int row  = tile * 16 + lo;

    const float* __restrict__ arow = nf + (size_t)row * D_NODE;
    v8f acc0 = {}, acc1 = {}, acc2 = {}, acc3 = {};
#pragma unroll 2
    for (int k0 = 0; k0 < D_NODE; k0 += 4) {
        v2f a;
        a.x = arow[k0 + 2 * hi];
        a.y = arow[k0 + 2 * hi + 1];
        const int r0 = (k0 + 2 * hi) * D_HID + lo;       // B row K=k0+2hi, col lo
        v2f b0, b1, b2, b3;
        b0.x = sW[r0];      b0.y = sW[r0 + D_HID];
        b1.x = sW[r0 + 16]; b1.y = sW[r0 + D_HID + 16];
        b2.x = sW[r0 + 32]; b2.y = sW[r0 + D_HID + 32];
        b3.x = sW[r0 + 48]; b3.y = sW[r0 + D_HID + 48];
        acc0 = __builtin_amdgcn_wmma_f32_16x16x4_f32(false, a, false, b0, (short)0, acc0, false, false);
        acc1 = __builtin_amdgcn_wmma_f32_16x16x4_f32(false, a, false, b1, (short)0, acc1, false, false);
        acc2 = __builtin_amdgcn_wmma_f32_16x16x4_f32(false, a, false, b2, (short)0, acc2, false, false);
        acc3 = __builtin_amdgcn_wmma_f32_16x16x4_f32(false, a, false, b3, (short)0, acc3, false, false);
    }

    // epilogue: bias + store, col tile t covers columns 16t..16t+15
    float* __restrict__ hrowbase = h + (size_t)(tile * 16 + 8 * hi) * D_HID + lo;
#pragma unroll
    for (int t = 0; t < 4; ++t) {
        const v8f acc = t == 0 ? acc0 : t == 1 ? acc1 : t == 2 ? acc2 : acc3;
        const float bias = wb[16 * t + lo];
#pragma unroll
        for (int v = 0; v < 8; ++v) {
            hrowbase[(size_t)v * D_HID + 16 * t] = acc[v] + bias;
        }
    }
}

// ---------------------------------------------------------------------------
// k_node_scores: s1[n] = h[n]·a_w[0:64], s2[n] = h[n]·a_w[64:128].
// One wave per node; lane handles 2 elements; shuffle-xor reduction.
// ---------------------------------------------------------------------------
__global__ __launch_bounds__(256) void k_node_scores(const float* __restrict__ h,
                                                     const float* __restrict__ aw,
                                                     float* __restrict__ s1,
                                                     float* __restrict__ s2) {
    long long gw = ((long long)blockIdx.x * blockDim.x + threadIdx.x) >> 5;
    int lane = threadIdx.x & 31;
    if (gw >= N_NODES) return;
    float2 hv = ((const float2*)(h + (size_t)gw * D_HID))[lane];
    float p1 = hv.x * aw[2 * lane]         + hv.y * aw[2 * lane + 1];
    float p2 = hv.x * aw[D_HID + 2 * lane] + hv.y * aw[D_HID + 2 * lane + 1];
#pragma unroll
    for (int off = 16; off > 0; off >>= 1) {
        p1 += __shfl_xor(p1, off, 32);
        p2 += __shfl_xor(p2, off, 32);
    }
    if (lane == 0) { s1[gw] = p1; s2[gw] = p2; }
}

// ---------------------------------------------------------------------------
// k_edge_scores: 4 lanes per edge. Lane quad dots 32 edge features with ve,
// reduces, adds gathered s1[tgt] + s2[nbr] + c, leaky-relu, stores eij and
// does atomic segment-max on monotone uint keys. edge_features is the largest
// HBM stream (164 MB) -> distance-prefetch it into GL2.
// ---------------------------------------------------------------------------
__global__ __launch_bounds__(256) void k_edge_scores(const float* __restrict__ ef,
                                                     const long long* __restrict__ ei,
                                                     const float* __restrict__ s1,
                                                     const float* __restrict__ s2,
                                                     const float* __restrict__ ve,
                                                     const float* __restrict__ cptr,
                                                     float* __restrict__ eij,
                                                     unsigned* __restrict__ keys) {
    long long t = (long long)blockIdx.x * blockDim.x + threadIdx.x;
    long long e = t >> 2;
    int q = (int)(t & 3);
    if (e >= N_EDGES) return;

    const float4* efe = (const float4*)(ef + e * D_EDGE);
    __builtin_prefetch(efe + 512, 0, 1);                 // 8 KB ahead (global_prefetch_b8)
    float4 x0 = efe[q * 2], x1 = efe[q * 2 + 1];
    const float4* vev = (const float4*)ve;
    float4 v0 = vev[q * 2], v1 = vev[q * 2 + 1];
    float p = x0.x * v0.x + x0.y * v0.y + x0.z * v0.z + x0.w * v0.w
            + x1.x * v1.x + x1.y * v1.y + x1.z * v1.z + x1.w * v1.w;
    p += __shfl_xor(p, 1, 32);
    p += __shfl_xor(p, 2, 32);

    if (q == 0) {
        int tgt = (int)ei[e];
        int nbr = (int)ei[N_EDGES + e];
        float sc = s1[tgt] + s2[nbr] + p + cptr[0];
        sc = sc >= 0.f ? sc : NEG_SLOPE * sc;
        eij[e] = sc;
        unsigned u = __float_as_uint(sc);
        unsigned key = (u & 0x80000000u) ? ~u : (u | 0x80000000u);
        atomicMax(&keys[tgt], key);
    }
}

// ---------------------------------------------------------------------------
// k_exp: ex = exp(eij - segmax[tgt]) (in-place over eij), atomic segment sum.
// ---------------------------------------------------------------------------
__global__ void k_exp(const long long* __restrict__ ei, float* __restrict__ eij,
                      const unsigned* __restrict__ keys, float* __restrict__ segsum) {
    long long e = (long long)blockIdx.x * blockDim.x + threadIdx.x;
    if (e >= N_EDGES) return;
    int tgt = (int)ei[e];
    unsigned key = keys[tgt];
    unsigned u = (key & 0x80000000u) ? (key ^ 0x80000000u) : ~key;
    float m = __uint_as_float(u);
    float ex = __expf(eij[e] - m);
    eij[e] = ex;
    atomicAdd(&segsum[tgt], ex);
}

// ---------------------------------------------------------------------------
// k_agg: out[tgt] += (ex/segsum[tgt]) * h[nbr].  Wave per edge, lane = 2 cols,
// global_atomic_add_f32 scatter (h and out are L2-resident: ~40 MB total).
// ---------------------------------------------------------------------------
__global__ __launch_bounds__(256) void k_agg(const long long* __restrict__ ei,
                                             const float* __restrict__ ex,
                                             const float* __restrict__ segsum,
                                             const float* __restrict__ h,
                                             float* __restrict__ out) {
    long long gw = ((long long)blockIdx.x * blockDim.x + threadIdx.x) >> 5;
    int lane = threadIdx.x & 31;
    if (gw >= N_EDGES) return;
    int tgt = (int)ei[gw];
    int nbr = (int)ei[N_EDGES + gw];
    float alpha = ex[gw] / segsum[tgt];
    float2 hv = ((const float2*)(h + (size_t)nbr * D_HID))[lane];
    atomicAdd(&out[(size_t)tgt * D_HID + 2 * lane],     alpha * hv.x);
    atomicAdd(&out[(size_t)tgt * D_HID + 2 * lane + 1], alpha * hv.y);
}

// ---------------------------------------------------------------------------
// k_elu: in-place ELU on the aggregated output.
// ---------------------------------------------------------------------------
__global__ void k_elu(float* __restrict__ out) {
    int i = blockIdx.x * blockDim.x + threadIdx.x;
    if (i >= N_NODES * D_HID) return;
    float x = out[i];
    out[i] = x > 0.f ? x : (__expf(x) - 1.f);
}

// ---------------------------------------------------------------------------
extern "C" void kernel_launch(void* const* d_in, const int* in_sizes, int n_in,
                              void* d_out, int out_size, void* d_ws, size_t ws_size,
                              hipStream_t stream) {
    const float*     nf  = (const float*)d_in[0];      // [N, 128]
    const float*     ef  = (const float*)d_in[1];      // [E, 32]
    const long long* ei  = (const long long*)d_in[2];  // [2, E] int64
    const float*     ww  = (const float*)d_in[3];      // [128, 64]
    const float*     wb  = (const float*)d_in[4];      // [64]
    const float*     eww = (const float*)d_in[5];      // [32, 64]
    const float*     ewb = (const float*)d_in[6];      // [64]
    const float*     aw  = (const float*)d_in[7];      // [192, 1]
    const float*     ab  = (const float*)d_in[8];      // [1]
    float* out = (float*)d_out;                        // [N, 64]

    // workspace layout (bytes)
    char* ws = (char*)d_ws;
    float*    h      = (float*)(ws);                            // 20,480,000 B
    float*    s1     = (float*)(ws + 20480000);                 //    320,000 B
    float*    s2     = (float*)(ws + 20800000);                 //    320,000 B
    float*    eij    = (float*)(ws + 21120000);                 //  5,120,000 B (reused as ex)
    unsigned* keys   = (unsigned*)(ws + 26240000);              //    320,000 B
    float*    segsum = (float*)(ws + 26560000);                 //    320,000 B
    float*    ve     = (float*)(ws + 26880000);                 //        128 B
    float*    cptr   = (float*)(ws + 26880128);                 //          4 B

    k_prep<<<1, 64, 0, stream>>>(eww, ewb, aw, ab, ve, cptr);

    k_init<<<(N_NODES * D_HID + 255) / 256, 256, 0, stream>>>(out, segsum, keys);

    // 80000/16 = 5000 row tiles, 8 waves/block -> 625 blocks (exact, EXEC all-1s)
    k_node_gemm<<<625, 256, 0, stream>>>(nf, ww, wb, h);

    // one wave per node: 80000 waves -> 10000 blocks
    k_node_scores<<<10000, 256, 0, stream>>>(h, aw, s1, s2);

    // 4 lanes/edge: 5.12M threads -> 20000 blocks
    k_edge_scores<<<(N_EDGES * 4 + 255) / 256, 256, 0, stream>>>(ef, ei, s1, s2, ve,
                                                                 cptr, eij, keys);

    k_exp<<<(N_EDGES + 255) / 256, 256, 0, stream>>>(ei, eij, keys, segsum);

    // wave per edge: 40.96M threads -> 160000 blocks
    k_agg<<<(int)(((long long)N_EDGES * 32 + 255) / 256), 256, 0, stream>>>(ei, eij,
                                                                            segsum, h, out);

    k_elu<<<(N_NODES * D_HID + 255) / 256, 256, 0, stream>>>(out);
}